// GlobalPointCloudOptimizer_68315749810628
// MI455X (gfx1250) — compile-verified
//
#include <hip/hip_runtime.h>
#include <hip/hip_bf16.h>
#include <math.h>

// MI455X / gfx1250. Memory-bound problem (~70MB effective traffic, L2-resident
// working set). WMMA f32 16x16x4 used as a Gram-matrix engine for the combo
// reductions: D = F F^T with F = 7xN feature matrix; A and B=A^T share the
// same per-lane registers under the documented 32-bit A/B VGPR layouts.

typedef __attribute__((ext_vector_type(2))) float v2f;
typedef __attribute__((ext_vector_type(8))) float v8f;

#define TILE  128     // points per wave-tile
#define FSTR  130     // LDS row stride (TILE + 2 pad -> bank spread)
#define NW    4       // waves per block in k_reduce
#define MAXPP 16

__device__ __forceinline__ void combo_fs(int c, int PP, int& f, int& s) {
  int f0 = 0, rem = c, cnt = PP - 1;
  while (rem >= cnt) { rem -= cnt; ++f0; --cnt; }
  f = f0; s = f0 + 1 + rem;
}

// ---------------------------------------------------------------- K0: logs
__global__ void k_logw(const float* __restrict__ ci, const float* __restrict__ cj,
                       float* __restrict__ wi, float* __restrict__ wj, int n) {
  int i = blockIdx.x * blockDim.x + threadIdx.x;
  if (i < n) { wi[i] = logf(ci[i]); wj[i] = logf(cj[i]); }
}

// ------------------------------------------- K1: per-combo Gram via WMMA
// grid = (2*NC, S), block = NW*32. Each block: one (scene, combo, side).
__global__ void k_reduce(const float* __restrict__ pts_i, const float* __restrict__ pts_j,
                         const float* __restrict__ wlogi, const float* __restrict__ wlogj,
                         const float* __restrict__ poses,
                         const int* __restrict__ v1, const int* __restrict__ v2,
                         float* __restrict__ sums, int N, int PP, int NC) {
  const int scene = blockIdx.y;
  const int side  = blockIdx.x & 1;
  const int c     = blockIdx.x >> 1;
  int f, s; combo_fs(c, PP, f, s);
  const int pf = scene * PP + f, ps = scene * PP + s;
  float* out = sums + ((size_t)(scene * NC + c) * 2 + side) * 6;
  const int tid = threadIdx.x, wv = tid >> 5, lane = tid & 31;

  const bool mask = (side == 0) ? (v1[pf] == v2[ps]) : (v2[pf] == v1[ps]);
  if (!mask) { if (tid < 6) out[tid] = 0.f; return; }

  const int iA = v1[pf], iB = v1[ps];
  float RA[9], RB[9];
  #pragma unroll
  for (int r = 0; r < 3; ++r)
    #pragma unroll
    for (int cc = 0; cc < 3; ++cc) {
      RA[r * 3 + cc] = poses[iA * 16 + r * 4 + cc];
      RB[r * 3 + cc] = poses[iB * 16 + r * 4 + cc];
    }
  const float t0 = poses[iB * 16 + 3]  - poses[iA * 16 + 3];
  const float t1 = poses[iB * 16 + 7]  - poses[iA * 16 + 7];
  const float t2 = poses[iB * 16 + 11] - poses[iA * 16 + 11];

  const float* wA = (side == 0) ? (wlogi + (size_t)pf * N) : (wlogj + (size_t)pf * N);
  const float* wB = (side == 0) ? (wlogj + (size_t)ps * N) : (wlogi + (size_t)ps * N);
  const float* pA = (side == 0) ? (pts_i + (size_t)pf * N * 3) : (pts_j + (size_t)pf * N * 3);
  const float* pB = (side == 0) ? (pts_j + (size_t)ps * N * 3) : (pts_i + (size_t)ps * N * 3);

  __shared__ float F[NW][16][FSTR];
  __shared__ float red[NW][6];
  for (int i = tid; i < NW * 16 * FSTR; i += blockDim.x) ((float*)F)[i] = 0.f;
  __syncthreads();

  v8f acc = {};
  const int ntiles = (N + TILE - 1) / TILE;
  const int ntp    = ((ntiles + NW - 1) / NW) * NW;
  const int koff   = (lane >> 4) << 1;   // A/B K offset: {0,2}
  const int feat   = lane & 15;          // A row / B col

  for (int t = wv; t < ntp; t += NW) {
    const int base = t * TILE;
    #pragma unroll
    for (int q = 0; q < TILE / 32; ++q) {
      const int pl = lane + q * 32;
      const int p  = base + pl;
      float w = 0.f, d1x = 0.f, d1y = 0.f, d1z = 0.f, d2x = 0.f, d2y = 0.f, d2z = 0.f;
      if (p < N) {
        const float ax = pA[(size_t)p * 3 + 0], ay = pA[(size_t)p * 3 + 1], az = pA[(size_t)p * 3 + 2];
        const float bx = pB[(size_t)p * 3 + 0], by = pB[(size_t)p * 3 + 1], bz = pB[(size_t)p * 3 + 2];
        d1x = RA[0] * ax + RA[1] * ay + RA[2] * az;
        d1y = RA[3] * ax + RA[4] * ay + RA[5] * az;
        d1z = RA[6] * ax + RA[7] * ay + RA[8] * az;
        d2x = RB[0] * bx + RB[1] * by + RB[2] * bz;
        d2y = RB[3] * bx + RB[4] * by + RB[5] * bz;
        d2z = RB[6] * bx + RB[7] * by + RB[8] * bz;
        w = fminf(wA[p], wB[p]);
      }
      F[wv][0][pl] = w * d1x; F[wv][1][pl] = w * d1y; F[wv][2][pl] = w * d1z;
      F[wv][3][pl] = w * d2x; F[wv][4][pl] = w * d2y; F[wv][5][pl] = w * d2z;
      F[wv][6][pl] = w;
    }
    // hint the next tile of this wave into cache (global_prefetch_b8)
    if (base + NW * TILE < N) {
      __builtin_prefetch(pA + (size_t)(base + NW * TILE) * 3, 0, 3);
      __builtin_prefetch(pB + (size_t)(base + NW * TILE) * 3, 0, 3);
    }
    __syncthreads();
    // Gram accumulate: D += A * A^T, K=4 points per wmma. A and B share regs.
    #pragma unroll
    for (int k4 = 0; k4 < TILE / 4; ++k4) {
      const v2f ab = *(const v2f*)&F[wv][feat][k4 * 4 + koff];
      acc = __builtin_amdgcn_wmma_f32_16x16x4_f32(false, ab, false, ab,
                                                  (short)0, acc, false, false);
    }
    __syncthreads();
  }

  // Extract needed D entries (vgpr m, lane n for m,n <= 6)
  const float sff = __shfl(acc[0], 0, 32) + __shfl(acc[1], 1, 32) + __shfl(acc[2], 2, 32);
  const float sfs = __shfl(acc[0], 3, 32) + __shfl(acc[1], 4, 32) + __shfl(acc[2], 5, 32);
  const float sss = __shfl(acc[3], 3, 32) + __shfl(acc[4], 4, 32) + __shfl(acc[5], 5, 32);
  const float s1x = __shfl(acc[6], 0, 32), s1y = __shfl(acc[6], 1, 32), s1z = __shfl(acc[6], 2, 32);
  const float s2x = __shfl(acc[6], 3, 32), s2y = __shfl(acc[6], 4, 32), s2z = __shfl(acc[6], 5, 32);
  const float sw2 = __shfl(acc[6], 6, 32);
  if (lane == 0) {
    red[wv][0] = sff;
    red[wv][1] = sfs;
    red[wv][2] = sss;
    red[wv][3] = s1x * t0 + s1y * t1 + s1z * t2;           // hf
    red[wv][4] = s2x * t0 + s2y * t1 + s2z * t2;           // hs
    red[wv][5] = sw2 * (t0 * t0 + t1 * t1 + t2 * t2);      // btb part
  }
  __syncthreads();
  if (tid < 6) {
    float v = 0.f;
    #pragma unroll
    for (int w = 0; w < NW; ++w) v += red[w][tid];
    out[tid] = v;
  }
}

// -------------------------------------- K2: assemble G,h; solve; residual
__global__ void k_solve(const float* __restrict__ sums, float* __restrict__ xsol,
                        float* __restrict__ vf, float* __restrict__ resid,
                        int PP, int NC) {
  if (threadIdx.x != 0) return;
  const int sc = blockIdx.x;
  const int M = PP;
  float G0[MAXPP][MAXPP], A[MAXPP][MAXPP], h0[MAXPP], b[MAXPP], x[MAXPP];
  for (int i = 0; i < M; ++i) { h0[i] = 0.f; for (int j = 0; j < M; ++j) G0[i][j] = 0.f; }
  float btbs = 0.f;
  for (int c = 0; c < NC; ++c) {
    int f, s; combo_fs(c, PP, f, s);
    const float* a = sums + (size_t)(sc * NC + c) * 12;
    const float sff = a[0] + a[6], sfs = a[1] + a[7], sss = a[2] + a[8];
    const float hf = a[3] + a[9], hs = a[4] + a[10], bt = a[5] + a[11];
    G0[f][f] += sff; G0[s][s] += sss; G0[f][s] -= sfs; G0[s][f] -= sfs;
    h0[f] += hf; h0[s] -= hs; btbs += bt;
  }
  for (int i = 0; i < M; ++i) { b[i] = h0[i]; for (int j = 0; j < M; ++j) A[i][j] = G0[i][j]; }
  // Gaussian elimination with partial pivoting
  for (int k = 0; k < M; ++k) {
    int piv = k; float best = fabsf(A[k][k]);
    for (int r = k + 1; r < M; ++r) { float v = fabsf(A[r][k]); if (v > best) { best = v; piv = r; } }
    if (piv != k) {
      for (int cc = k; cc < M; ++cc) { float tp = A[k][cc]; A[k][cc] = A[piv][cc]; A[piv][cc] = tp; }
      float tb = b[k]; b[k] = b[piv]; b[piv] = tb;
    }
    float d = A[k][k];
    if (fabsf(d) < 1e-30f) d = (d < 0.f ? -1e-30f : 1e-30f);
    const float inv = 1.f / d;
    for (int r = k + 1; r < M; ++r) {
      const float fct = A[r][k] * inv;
      if (fct != 0.f) {
        for (int cc = k; cc < M; ++cc) A[r][cc] -= fct * A[k][cc];
        b[r] -= fct * b[k];
      }
    }
  }
  for (int i = M - 1; i >= 0; --i) {
    float ss = b[i];
    for (int j = i + 1; j < M; ++j) ss -= A[i][j] * x[j];
    float d = A[i][i];
    if (fabsf(d) < 1e-30f) d = (d < 0.f ? -1e-30f : 1e-30f);
    x[i] = ss / d;
  }
  float res = btbs, mn = x[0];
  for (int i = 0; i < M; ++i) {
    float gx = 0.f;
    for (int j = 0; j < M; ++j) gx += G0[i][j] * x[j];
    res += x[i] * gx - 2.f * x[i] * h0[i];
    mn = fminf(mn, x[i]);
  }
  vf[sc] = (mn > 0.f) ? 1.f : 0.f;
  resid[sc] = res;
  for (int i = 0; i < M; ++i) xsol[sc * PP + i] = x[i];
}

// ------------- K3: res_mean, valid, zm, per-pair fused params, pair lists
__global__ void k_finalize(const float* __restrict__ xsol, const float* __restrict__ vf,
                           const float* __restrict__ resid, const float* __restrict__ poses,
                           const int* __restrict__ v1, const int* __restrict__ v2,
                           float* __restrict__ zm, float* __restrict__ pprm,
                           int* __restrict__ lists, float* __restrict__ dout,
                           int S, int PP, int NPAIR, int NIMG) {
  const int t = threadIdx.x;
  if (t == 0) {
    float num = 0.f, den = 0.f;
    for (int s = 0; s < S; ++s) { num += resid[s] * vf[s]; den += vf[s]; }
    dout[0] = num / fmaxf(den, 1.f);
  }
  if (t < S) dout[1 + t] = vf[t];
  for (int img = t; img < NIMG; img += blockDim.x) {
    float z = 0.f; int c1 = 0, c2 = 0;
    int* L = lists + img * 34;
    for (int p = 0; p < NPAIR; ++p) {
      if (v1[p] == img) { z += vf[p / PP]; if (c1 < 16) L[1 + c1++] = p; }
      if (v2[p] == img) { if (c2 < 16) L[18 + c2++] = p; }
    }
    L[0] = c1; L[17] = c2; zm[img] = z;
  }
  for (int p = t; p < NPAIR; p += blockDim.x) {
    const int sc = p / PP;
    const float vfp = vf[sc];
    const float se = xsol[sc * PP + (p % PP)] * vfp;   // s_eff (scales * vpair)
    const int i1 = v1[p], i2 = v2[p];
    float R1m[9], R2m[9], T1[3], T2[3];
    for (int r = 0; r < 3; ++r) {
      for (int cc = 0; cc < 3; ++cc) {
        R1m[r * 3 + cc] = poses[i1 * 16 + r * 4 + cc];
        R2m[r * 3 + cc] = poses[i2 * 16 + r * 4 + cc];
      }
      T1[r] = poses[i1 * 16 + r * 4 + 3];
      T2[r] = poses[i2 * 16 + r * 4 + 3];
    }
    float* q = pprm + (size_t)p * 14;
    // M = se * R2^T R1 ; b = R2^T (T1 - T2)
    for (int a = 0; a < 3; ++a) {
      for (int bb = 0; bb < 3; ++bb)
        q[a * 3 + bb] = se * (R2m[0 * 3 + a] * R1m[0 * 3 + bb] +
                              R2m[1 * 3 + a] * R1m[1 * 3 + bb] +
                              R2m[2 * 3 + a] * R1m[2 * 3 + bb]);
      q[9 + a] = R2m[0 * 3 + a] * (T1[0] - T2[0]) +
                 R2m[1 * 3 + a] * (T1[1] - T2[1]) +
                 R2m[2 * 3 + a] * (T1[2] - T2[2]);
    }
    q[12] = se; q[13] = vfp;
  }
}

// ---------------------------------------------------------- K4: cam blend
__global__ void k_cam(const float* __restrict__ pts_i, const float* __restrict__ pts_j,
                      const float* __restrict__ wlogi, const float* __restrict__ wlogj,
                      const float* __restrict__ zm, const float* __restrict__ pprm,
                      const int* __restrict__ lists, float* __restrict__ cam,
                      int N, int P) {
  const int img = blockIdx.y;
  const int n = blockIdx.x * blockDim.x + threadIdx.x;
  if (n >= N) return;
  const int* L = lists + img * 34;
  float ax, ay, az;
  if (zm[img] > 0.f) { ax = ay = az = 0.f; }
  else {
    const float* ip = pts_i + ((size_t)img * (P - 1) * N + n) * 3;
    ax = ip[0]; ay = ip[1]; az = ip[2];
  }
  float den = 0.f;
  const int c1 = L[0];
  for (int k = 0; k < c1; ++k) {
    const int p = L[1 + k];
    const float* q = pprm + (size_t)p * 14;
    const float w = wlogi[(size_t)p * N + n] * q[13];
    const float se = q[12];
    const float* pp = pts_i + ((size_t)p * N + n) * 3;
    ax += se * pp[0] * w; ay += se * pp[1] * w; az += se * pp[2] * w;
    den += w;
  }
  const int c2 = L[17];
  for (int k = 0; k < c2; ++k) {
    const int p = L[18 + k];
    const float* q = pprm + (size_t)p * 14;
    const float w = wlogj[(size_t)p * N + n] * q[13];
    const float* pp = pts_j + ((size_t)p * N + n) * 3;
    const float vx = q[0] * pp[0] + q[1] * pp[1] + q[2] * pp[2] + q[9];
    const float vy = q[3] * pp[0] + q[4] * pp[1] + q[5] * pp[2] + q[10];
    const float vz = q[6] * pp[0] + q[7] * pp[1] + q[8] * pp[2] + q[11];
    ax += vx * w; ay += vy * w; az += vz * w;
    den += w;
  }
  if (den > 0.f) { ax /= den; ay /= den; az /= den; }
  float* o = cam + ((size_t)img * N + n) * 3;
  o[0] = ax; o[1] = ay; o[2] = az;
}

extern "C" void kernel_launch(void* const* d_in, const int* in_sizes, int n_in,
                              void* d_out, int out_size, void* d_ws, size_t ws_size,
                              hipStream_t stream) {
  const float* pts_i  = (const float*)d_in[0];
  const float* pts_j  = (const float*)d_in[1];
  const float* conf_i = (const float*)d_in[2];
  const float* conf_j = (const float*)d_in[3];
  const float* poses  = (const float*)d_in[4];
  const int*   v1     = (const int*)d_in[5];
  const int*   v2     = (const int*)d_in[6];

  const int NPAIR = in_sizes[5];
  const int NIMG  = in_sizes[4] / 16;
  const int N     = in_sizes[2] / NPAIR;
  const int P     = NPAIR / NIMG + 1;
  const int PP    = P * (P - 1);
  const int S     = NIMG / P;
  const int NC    = PP * (PP - 1) / 2;

  float* ws    = (float*)d_ws;
  float* wlogi = ws;
  float* wlogj = wlogi + (size_t)NPAIR * N;
  float* sums  = wlogj + (size_t)NPAIR * N;
  float* xsol  = sums + (size_t)S * NC * 12;
  float* vf    = xsol + (size_t)S * PP;
  float* resid = vf + S;
  float* zmv   = resid + S;
  float* pprm  = zmv + NIMG;
  int*   lists = (int*)(pprm + (size_t)NPAIR * 14);

  float* dout = (float*)d_out;
  float* cam  = dout + 1 + S;

  const int tot = NPAIR * N;
  k_logw<<<(tot + 255) / 256, 256, 0, stream>>>(conf_i, conf_j, wlogi, wlogj, tot);
  k_reduce<<<dim3(2 * NC, S), NW * 32, 0, stream>>>(pts_i, pts_j, wlogi, wlogj,
                                                    poses, v1, v2, sums, N, PP, NC);
  k_solve<<<S, 32, 0, stream>>>(sums, xsol, vf, resid, PP, NC);
  k_finalize<<<1, 128, 0, stream>>>(xsol, vf, resid, poses, v1, v2,
                                    zmv, pprm, lists, dout, S, PP, NPAIR, NIMG);
  k_cam<<<dim3((N + 255) / 256, NIMG), 256, 0, stream>>>(pts_i, pts_j, wlogi, wlogj,
                                                         zmv, pprm, lists, cam, N, P);
}